// TCN_GCN_unit_36661840838903
// MI455X (gfx1250) — compile-verified
//
#include <hip/hip_runtime.h>
#include <hip/hip_bf16.h>
#include <math.h>

// ---- problem dims (from reference) ----
#define N_   32
#define C_   96
#define T_   256
#define V_   25
#define S_   3
#define REL_ 12
#define OUT_ 96
#define BC_  24
#define KS_  5

// ---- GEMM tiling ----
#define TT     8            // t-tile
#define MROWS  (TT * V_)    // 200 valid rows (t,v)
#define MT     13           // 13 x 16 = 208 padded rows
#define MPAD   (MT * 16)    // 208 (keeps ds_load_2addr offset1 in range)

typedef float v2f __attribute__((ext_vector_type(2)));
typedef float v8f __attribute__((ext_vector_type(8)));

__device__ __forceinline__ v8f wmma4(v2f a, v2f b, v8f c) {
  // D = A(16x4 f32) x B(4x16 f32) + C(16x16 f32)  -> v_wmma_f32_16x16x4_f32
  return __builtin_amdgcn_wmma_f32_16x16x4_f32(false, a, false, b, (short)0, c,
                                               false, false);
}

// -------------------------------------------------------------------------
// K1: xm[n,c,v] = mean_t x[n,c,t,v]
// -------------------------------------------------------------------------
__global__ void k_mean(const float* __restrict__ x, float* __restrict__ xm) {
  int idx = blockIdx.x * blockDim.x + threadIdx.x;
  if (idx >= N_ * C_ * V_) return;
  int v = idx % V_;
  int nc = idx / V_;
  const float* p = x + (size_t)nc * T_ * V_ + v;
  float s = 0.f;
#pragma unroll 8
  for (int t = 0; t < T_; ++t) s += p[(size_t)t * V_];
  xm[idx] = s * (1.f / (float)T_);
}

// -------------------------------------------------------------------------
// K2: A_at[n,s,o,u,v] = alpha*(sum_r tanh(x1[r,u]-x2[r,v])*c4w[s,o,r]+c4b)
//                       + A3[s,u,v] + beta*A6[o%6,u,v] + gamma*spd[u,v]
// one block per (n,s)
// -------------------------------------------------------------------------
__global__ void __launch_bounds__(256) k_adapt(
    const float* __restrict__ xm, const float* __restrict__ spd,
    const float* __restrict__ A3, const float* __restrict__ A6,
    const float* __restrict__ alpha, const float* __restrict__ beta,
    const float* __restrict__ gamma,
    const float* __restrict__ c1w, const float* __restrict__ c1b,
    const float* __restrict__ c2w, const float* __restrict__ c2b,
    const float* __restrict__ c4w, const float* __restrict__ c4b,
    float* __restrict__ wsA) {
  int ns = blockIdx.x;
  int n = ns / S_, s = ns % S_;
  int tid = threadIdx.x;
  __shared__ float x1s[REL_ * V_];
  __shared__ float x2s[REL_ * V_];
  __shared__ float Rt[REL_ * V_ * V_];

  for (int e = tid; e < 2 * REL_ * V_; e += blockDim.x) {
    int which = e / (REL_ * V_);
    int rv = e % (REL_ * V_);
    int r = rv / V_, v = rv % V_;
    const float* w = (which ? c2w : c1w) + (size_t)(s * REL_ + r) * C_;
    const float* xp = xm + (size_t)n * C_ * V_ + v;
    float acc = 0.f;
    for (int c = 0; c < C_; ++c) acc += xp[(size_t)c * V_] * w[c];
    acc += (which ? c2b : c1b)[s * REL_ + r];
    (which ? x2s : x1s)[rv] = acc;
  }
  __syncthreads();
  for (int e = tid; e < REL_ * V_ * V_; e += blockDim.x) {
    int r = e / (V_ * V_);
    int uv = e % (V_ * V_);
    int u = uv / V_, v = uv % V_;
    Rt[e] = tanhf(x1s[r * V_ + u] - x2s[r * V_ + v]);
  }
  __syncthreads();
  float al = alpha[0], be = beta[0], ga = gamma[0];
  for (int e = tid; e < OUT_ * V_ * V_; e += blockDim.x) {
    int o = e / (V_ * V_);
    int uv = e % (V_ * V_);
    float m = c4b[s * OUT_ + o];
    const float* w = c4w + (size_t)(s * OUT_ + o) * REL_;
#pragma unroll
    for (int r = 0; r < REL_; ++r) m += Rt[r * V_ * V_ + uv] * w[r];
    float a = m * al + A3[(size_t)s * V_ * V_ + uv] +
              A6[(size_t)(o % 6) * V_ * V_ + uv] * be + spd[uv] * ga;
    wsA[(size_t)ns * OUT_ * V_ * V_ + (size_t)o * V_ * V_ + uv] = a;
  }
}

// -------------------------------------------------------------------------
// K3: fused unit_gcn main path.
//   stage1 (WMMA f32 16x16x4): x3[o',m] = sum_c x[n,c,m] * c3w[s,o,c] + b
//       rows m = t*25+v (200 valid, padded to 208), cols = 16 o's, K = 96
//       K staged through LDS in 3 chunks of 32; A/B fragments hoisted to
//       registers per chunk so the 8 WMMAs issue back-to-back; next chunk
//       prefetched via global_prefetch_b8.
//   stage2 (VALU): y[o',t,u] += sum_v A_at[n,s,o,u,v] * x3[o',t,v]
//   epilogue: y = relu(gbn_s*y + gbn_b + x)
// block = (otile, ttile, n), 256 threads (8 wave32)
// -------------------------------------------------------------------------
__global__ void __launch_bounds__(256) k_gcn(
    const float* __restrict__ x,
    const float* __restrict__ c3w, const float* __restrict__ c3b,
    const float* __restrict__ gbn_s, const float* __restrict__ gbn_b,
    const float* __restrict__ wsA, float* __restrict__ wsY) {
  int obase = blockIdx.x * 16;
  int t0 = blockIdx.y * TT;
  int n = blockIdx.z;
  int tid = threadIdx.x;
  int lane = tid & 31;
  int wave = tid >> 5;
  int half = (lane >> 4) & 1;  // K sub-pair (0 -> K0,1 ; 1 -> K2,3)
  int l15 = lane & 15;

  __shared__ float xs[32][MPAD];    // K-chunk of A rows
  __shared__ float x3s[16][MPAD];   // stage1 result [o'][m]
  __shared__ float wsm[16][C_];     // weight tile [o'][c]

  // stage-2 thread mapping: 16 o' x 8 t x 2 u-halves = 256
  int op = tid >> 4;
  int q = tid & 15;
  int t2 = q >> 1;
  int uhalf = q & 1;
  int u0 = uhalf ? 13 : 0;
  int nu = uhalf ? 12 : 13;
  float yacc[13];
#pragma unroll
  for (int i = 0; i < 13; ++i) yacc[i] = 0.f;

  for (int s = 0; s < S_; ++s) {
    __syncthreads();
    for (int e = tid; e < 16 * C_; e += 256) {
      int o = e / C_, c = e % C_;
      wsm[o][c] = c3w[(size_t)(s * OUT_ + obase + o) * C_ + c];
    }
    v8f acc0 = {0.f, 0.f, 0.f, 0.f, 0.f, 0.f, 0.f, 0.f};
    v8f acc1 = {0.f, 0.f, 0.f, 0.f, 0.f, 0.f, 0.f, 0.f};
    for (int kc = 0; kc < 3; ++kc) {
      __syncthreads();
      for (int e = tid; e < 32 * MPAD; e += 256) {
        int cl = e / MPAD, m = e % MPAD;
        float vl = 0.f;
        if (m < MROWS) {
          int t = m / V_, v = m % V_;
          size_t gi = (((size_t)n * C_ + kc * 32 + cl) * T_ + t0 + t) * V_ + v;
          vl = x[gi];
          if (kc < 2)  // prefetch next K-chunk (+32 channels)
            __builtin_prefetch(x + gi + (size_t)32 * T_ * V_, 0, 0);
        }
        xs[cl][m] = vl;
      }
      __syncthreads();
      // hoist all fragments for this chunk, then run the WMMA chain
      v2f bv[8], a0v[8];
      int m0 = wave * 16 + l15;
#pragma unroll
      for (int kk = 0; kk < 8; ++kk) {
        int kl = kk * 4 + half * 2;
        bv[kk].x = wsm[l15][kc * 32 + kl];
        bv[kk].y = wsm[l15][kc * 32 + kl + 1];
        a0v[kk].x = xs[kl][m0];
        a0v[kk].y = xs[kl + 1][m0];
      }
#pragma unroll
      for (int kk = 0; kk < 8; ++kk) acc0 = wmma4(a0v[kk], bv[kk], acc0);
      if (wave < MT - 8) {  // wave-uniform: EXEC stays all-ones for WMMA
        int m1 = m0 + 8 * 16;
        v2f a1v[8];
#pragma unroll
        for (int kk = 0; kk < 8; ++kk) {
          int kl = kk * 4 + half * 2;
          a1v[kk].x = xs[kl][m1];
          a1v[kk].y = xs[kl + 1][m1];
        }
#pragma unroll
        for (int kk = 0; kk < 8; ++kk) acc1 = wmma4(a1v[kk], bv[kk], acc1);
      }
    }
    __syncthreads();
    // D layout: VGPR r -> row (mt*16 + r + half*8), col l15
    float bias = c3b[s * OUT_ + obase + l15];
#pragma unroll
    for (int r = 0; r < 8; ++r) {
      int m = wave * 16 + half * 8 + r;
      if (m < MROWS) x3s[l15][m] = acc0[r] + bias;
    }
    if (wave < MT - 8) {
#pragma unroll
      for (int r = 0; r < 8; ++r) {
        int m = (8 + wave) * 16 + half * 8 + r;
        if (m < MROWS) x3s[l15][m] = acc1[r] + bias;
      }
    }
    __syncthreads();
    // stage 2: 25x25 graph mixing per (n, o)
    {
      const float* Abase =
          wsA + ((size_t)(n * S_ + s) * OUT_ + obase + op) * (V_ * V_);
      const float* xr = &x3s[op][t2 * V_];
      for (int ui = 0; ui < nu; ++ui) {
        int u = u0 + ui;
        const float* av = Abase + u * V_;
        float acc = 0.f;
#pragma unroll
        for (int v = 0; v < V_; ++v) acc += av[v] * xr[v];
        yacc[ui] += acc;
      }
    }
  }
  // epilogue: BN + residual + relu
  {
    int o = obase + op;
    float gs = gbn_s[o], gb = gbn_b[o];
    size_t base = (((size_t)n * OUT_ + o) * T_ + t0 + t2) * V_;
    for (int ui = 0; ui < nu; ++ui) {
      int u = u0 + ui;
      float val = gs * yacc[ui] + gb + x[base + u];
      wsY[base + u] = fmaxf(val, 0.f);
    }
  }
}

// -------------------------------------------------------------------------
// K4: all 4 pointwise branches fused as one WMMA GEMM (pw_w is [96][96]
// contiguous), per-channel BN, ReLU for branches 0..2 only.
// -------------------------------------------------------------------------
__global__ void __launch_bounds__(256) k_pw(
    const float* __restrict__ wsY,
    const float* __restrict__ pw_w, const float* __restrict__ pw_b,
    const float* __restrict__ pwbn_s, const float* __restrict__ pwbn_b,
    float* __restrict__ wsZ) {
  int nbase = blockIdx.x * 16;
  int t0 = blockIdx.y * TT;
  int n = blockIdx.z;
  int tid = threadIdx.x;
  int lane = tid & 31;
  int wave = tid >> 5;
  int half = (lane >> 4) & 1;
  int l15 = lane & 15;

  __shared__ float xs[32][MPAD];
  __shared__ float wsm[16][C_];

  for (int e = tid; e < 16 * C_; e += 256) {
    int o = e / C_, c = e % C_;
    wsm[o][c] = pw_w[(size_t)(nbase + o) * C_ + c];
  }
  v8f acc0 = {0.f, 0.f, 0.f, 0.f, 0.f, 0.f, 0.f, 0.f};
  v8f acc1 = {0.f, 0.f, 0.f, 0.f, 0.f, 0.f, 0.f, 0.f};
  for (int kc = 0; kc < 3; ++kc) {
    __syncthreads();
    for (int e = tid; e < 32 * MPAD; e += 256) {
      int cl = e / MPAD, m = e % MPAD;
      float vl = 0.f;
      if (m < MROWS) {
        int t = m / V_, v = m % V_;
        size_t gi = (((size_t)n * C_ + kc * 32 + cl) * T_ + t0 + t) * V_ + v;
        vl = wsY[gi];
        if (kc < 2)
          __builtin_prefetch(wsY + gi + (size_t)32 * T_ * V_, 0, 0);
      }
      xs[cl][m] = vl;
    }
    __syncthreads();
    v2f bv[8], a0v[8];
    int m0 = wave * 16 + l15;
#pragma unroll
    for (int kk = 0; kk < 8; ++kk) {
      int kl = kk * 4 + half * 2;
      bv[kk].x = wsm[l15][kc * 32 + kl];
      bv[kk].y = wsm[l15][kc * 32 + kl + 1];
      a0v[kk].x = xs[kl][m0];
      a0v[kk].y = xs[kl + 1][m0];
    }
#pragma unroll
    for (int kk = 0; kk < 8; ++kk) acc0 = wmma4(a0v[kk], bv[kk], acc0);
    if (wave < MT - 8) {
      int m1 = m0 + 8 * 16;
      v2f a1v[8];
#pragma unroll
      for (int kk = 0; kk < 8; ++kk) {
        int kl = kk * 4 + half * 2;
        a1v[kk].x = xs[kl][m1];
        a1v[kk].y = xs[kl + 1][m1];
      }
#pragma unroll
      for (int kk = 0; kk < 8; ++kk) acc1 = wmma4(a1v[kk], bv[kk], acc1);
    }
  }
  int ch = nbase + l15;          // 0..95 = b*24+j
  float bias = pw_b[ch];
  float sc = pwbn_s[ch], sh = pwbn_b[ch];
  bool dorelu = (ch < 3 * BC_);  // branch 3 gets BN but no relu
  auto emit = [&](v8f& acc, int mt) {
#pragma unroll
    for (int r = 0; r < 8; ++r) {
      int m = mt * 16 + half * 8 + r;
      if (m < MROWS) {
        int t = m / V_, v = m % V_;
        float val = sc * (acc[r] + bias) + sh;
        if (dorelu) val = fmaxf(val, 0.f);
        wsZ[(((size_t)n * 96 + ch) * T_ + t0 + t) * V_ + v] = val;
      }
    }
  };
  emit(acc0, wave);
  if (wave < MT - 8) emit(acc1, 8 + wave);
}

// -------------------------------------------------------------------------
// K5: temporal convs (b=0 d=1, b=1 d=2), maxpool+BN (b=2), passthrough
// (b=3), concat + outer residual + relu.
// -------------------------------------------------------------------------
__global__ void k_tcn(const float* __restrict__ x, const float* __restrict__ wsZ,
                      const float* __restrict__ tc_w,
                      const float* __restrict__ tc_b,
                      const float* __restrict__ tcbn_s,
                      const float* __restrict__ tcbn_b,
                      const float* __restrict__ mpbn_s,
                      const float* __restrict__ mpbn_b,
                      float* __restrict__ out) {
  int idx = blockIdx.x * blockDim.x + threadIdx.x;
  if (idx >= N_ * OUT_ * T_ * V_) return;
  int v = idx % V_;
  int t = (idx / V_) % T_;
  int co = (idx / (V_ * T_)) % OUT_;
  int n = idx / (V_ * T_ * OUT_);
  int b = co / BC_, j = co % BC_;
  float val;
  if (b < 2) {
    int d = (b == 0) ? 1 : 2;
    float acc = tc_b[b * BC_ + j];
    const float* wbase = tc_w + (size_t)(b * BC_ + j) * BC_ * KS_;
#pragma unroll
    for (int k = 0; k < KS_; ++k) {
      int tt = t + (k - 2) * d;
      if ((unsigned)tt < (unsigned)T_) {
        const float* zp = wsZ + (((size_t)n * 96 + b * BC_) * T_ + tt) * V_ + v;
        for (int ic = 0; ic < BC_; ++ic)
          acc += zp[(size_t)ic * T_ * V_] * wbase[ic * KS_ + k];
      }
    }
    val = tcbn_s[b * BC_ + j] * acc + tcbn_b[b * BC_ + j];
  } else if (b == 2) {
    float m = -3.402823466e38f;
    for (int dt = -1; dt <= 1; ++dt) {
      int tt = t + dt;
      if ((unsigned)tt < (unsigned)T_)
        m = fmaxf(m, wsZ[(((size_t)n * 96 + 48 + j) * T_ + tt) * V_ + v]);
    }
    val = mpbn_s[j] * m + mpbn_b[j];
  } else {
    val = wsZ[(((size_t)n * 96 + 72 + j) * T_ + t) * V_ + v];
  }
  float r = val + x[idx];
  out[idx] = fmaxf(r, 0.f);
}

// -------------------------------------------------------------------------
extern "C" void kernel_launch(void* const* d_in, const int* in_sizes, int n_in,
                              void* d_out, int out_size, void* d_ws,
                              size_t ws_size, hipStream_t stream) {
  (void)in_sizes; (void)n_in; (void)out_size; (void)ws_size;
  const float* x      = (const float*)d_in[0];
  const float* spd    = (const float*)d_in[1];
  const float* A3     = (const float*)d_in[2];
  const float* A6     = (const float*)d_in[3];
  const float* alpha  = (const float*)d_in[4];
  const float* beta   = (const float*)d_in[5];
  const float* gamma  = (const float*)d_in[6];
  const float* c1w    = (const float*)d_in[7];
  const float* c1b    = (const float*)d_in[8];
  const float* c2w    = (const float*)d_in[9];
  const float* c2b    = (const float*)d_in[10];
  const float* c4w    = (const float*)d_in[11];
  const float* c4b    = (const float*)d_in[12];
  const float* c3w    = (const float*)d_in[13];
  const float* c3b    = (const float*)d_in[14];
  const float* gbn_s  = (const float*)d_in[15];
  const float* gbn_b  = (const float*)d_in[16];
  const float* pw_w   = (const float*)d_in[17];
  const float* pw_b   = (const float*)d_in[18];
  const float* pwbn_s = (const float*)d_in[19];
  const float* pwbn_b = (const float*)d_in[20];
  const float* tc_w   = (const float*)d_in[21];
  const float* tc_b   = (const float*)d_in[22];
  const float* tcbn_s = (const float*)d_in[23];
  const float* tcbn_b = (const float*)d_in[24];
  const float* mpbn_s = (const float*)d_in[25];
  const float* mpbn_b = (const float*)d_in[26];
  float* out = (float*)d_out;

  float* ws   = (float*)d_ws;
  float* wsXM = ws;                                     // N*C*V
  float* wsA  = wsXM + (size_t)N_ * C_ * V_;            // N*S*OUT*V*V
  float* wsY  = wsA + (size_t)N_ * S_ * OUT_ * V_ * V_; // N*OUT*T*V
  float* wsZ  = wsY + (size_t)N_ * OUT_ * T_ * V_;      // N*96*T*V

  k_mean<<<(N_ * C_ * V_ + 255) / 256, 256, 0, stream>>>(x, wsXM);
  k_adapt<<<N_ * S_, 256, 0, stream>>>(wsXM, spd, A3, A6, alpha, beta, gamma,
                                       c1w, c1b, c2w, c2b, c4w, c4b, wsA);
  dim3 g3(OUT_ / 16, T_ / TT, N_);
  k_gcn<<<g3, 256, 0, stream>>>(x, c3w, c3b, gbn_s, gbn_b, wsA, wsY);
  dim3 g4(96 / 16, T_ / TT, N_);
  k_pw<<<g4, 256, 0, stream>>>(wsY, pw_w, pw_b, pwbn_s, pwbn_b, wsZ);
  int total = N_ * OUT_ * T_ * V_;
  k_tcn<<<(total + 255) / 256, 256, 0, stream>>>(x, wsZ, tc_w, tc_b, tcbn_s,
                                                 tcbn_b, mpbn_s, mpbn_b, out);
}